// SelectiveSSM_15238543966730
// MI455X (gfx1250) — compile-verified
//
#include <hip/hip_runtime.h>
#include <math.h>

typedef __attribute__((ext_vector_type(2))) float v2f;
typedef __attribute__((ext_vector_type(8))) float v8f;

#define D_MODEL 1024
#define D_INNER 2048
#define D_STATE 16
#define DT_RANK 64
#define NXZ 96            // DT_RANK + 2*D_STATE
#define BATCH 2
#define SEQ 2048
#define NROWS (BATCH * SEQ)   // 4096 GEMM rows

// ---------------------------------------------------------------------------
// K1: xz[r][j] = sum_k x[r][k] * Wx[k][j]   (M=4096, K=2048, N=96)
// 256 blocks (16-row M tiles), 6 waves per block, one 16-col N tile per wave.
// Double-buffered LDS; Wx stored K-pair-packed so each B fragment is one
// aligned ds_load_b64 (no v_mov repacking in front of the WMMA).
// ---------------------------------------------------------------------------
__global__ __launch_bounds__(192)
void k_xz_gemm(const float* __restrict__ x, const float* __restrict__ Wx,
               float* __restrict__ xz) {
  __shared__ alignas(16) float ldsA[2][16 * 64];    // 2 x 4 KB
  __shared__ alignas(16) float ldsB[2][32 * 192];   // 2 x 24 KB, pair-packed:
                                                    // [kpair][n*2 + (k&1)]
  const int m0   = blockIdx.x * 16;
  const int tid  = threadIdx.x;
  const int wave = tid >> 5;
  const int lane = tid & 31;
  const int n0   = wave * 16;          // 6 waves cover N=96
  const int Mf   = lane & 15;          // row (A) / col (B) within tile
  const int K0   = (lane >> 4) * 2;    // K offset selected by lane[4]

  auto loadChunk = [&](int kc, int buf) {
    for (int idx = tid; idx < 16 * 64; idx += 192) {
      int r = idx >> 6, cl = idx & 63;
      ldsA[buf][idx] = x[(size_t)(m0 + r) * D_INNER + kc + cl];
    }
    for (int idx = tid; idx < 32 * 96; idx += 192) {   // 32 K-pairs x 96 cols
      int kp = idx / 96, nn = idx - kp * 96;
      ldsB[buf][kp * 192 + nn * 2]     = Wx[(size_t)(kc + 2 * kp) * NXZ + nn];
      ldsB[buf][kp * 192 + nn * 2 + 1] = Wx[(size_t)(kc + 2 * kp + 1) * NXZ + nn];
    }
  };

  loadChunk(0, 0);
  v8f c = {};
  for (int ci = 0; ci < D_INNER / 64; ++ci) {
    __syncthreads();
    if (ci + 1 < D_INNER / 64)
      loadChunk((ci + 1) * 64, (ci + 1) & 1);   // overlaps WMMAs below
    const float* A = ldsA[ci & 1];
    const float* B = ldsB[ci & 1];
    #pragma unroll
    for (int kk = 0; kk < 64; kk += 4) {
      v2f a = *(const v2f*)&A[Mf * 64 + kk + K0];                 // ds_load_b64
      v2f b = *(const v2f*)&B[((kk + K0) >> 1) * 192 + (n0 + Mf) * 2]; // ds_load_b64
      c = __builtin_amdgcn_wmma_f32_16x16x4_f32(false, a, false, b,
                                                (short)0, c, false, false);
    }
  }
  // C/D layout: vgpr r -> M = r + 8*lane[4], N = lane[3:0]
  const int nOut  = n0 + Mf;
  const int mBase = m0 + ((lane >> 4) << 3);
  #pragma unroll
  for (int r = 0; r < 8; ++r)
    xz[(size_t)(mBase + r) * NXZ + nOut] = c[r];
}

// ---------------------------------------------------------------------------
// K2: dt[r][d] = softplus( sum_k delta[r][k]*Wdt[k][d] + bdt[d] )
// M=4096, K=64, N=2048. grid (256 m-tiles, 16 n-groups of 128), 8 waves/block.
// delta tile (16x64) and pair-packed Wdt tile (64x128) staged in LDS once.
// ---------------------------------------------------------------------------
__global__ __launch_bounds__(256)
void k_dt_gemm(const float* __restrict__ xz, const float* __restrict__ Wdt,
               const float* __restrict__ bdt, float* __restrict__ dt) {
  __shared__ alignas(16) float ldsA[16 * 64];     // 4 KB
  __shared__ alignas(16) float ldsW[32 * 256];    // 32 KB, pair-packed Wdt tile
  const int m0   = blockIdx.x * 16;
  const int nb0  = blockIdx.y * 128;
  const int tid  = threadIdx.x;
  const int wave = tid >> 5;
  const int lane = tid & 31;

  for (int idx = tid; idx < 16 * 64; idx += 256) {
    int r = idx >> 6, cl = idx & 63;
    ldsA[idx] = xz[(size_t)(m0 + r) * NXZ + cl];  // delta_raw = xz cols 0..63
  }
  for (int idx = tid; idx < 32 * 128; idx += 256) {  // 32 K-pairs x 128 cols
    int kp = idx >> 7, nl = idx & 127;
    ldsW[kp * 256 + nl * 2]     = Wdt[(size_t)(2 * kp) * D_INNER + nb0 + nl];
    ldsW[kp * 256 + nl * 2 + 1] = Wdt[(size_t)(2 * kp + 1) * D_INNER + nb0 + nl];
  }
  __syncthreads();

  const int Mf = lane & 15;
  const int K0 = (lane >> 4) * 2;
  const int nl = wave * 16 + Mf;       // column within the 128-wide block tile
  v8f c = {};
  #pragma unroll
  for (int k = 0; k < DT_RANK; k += 4) {
    v2f a = *(const v2f*)&ldsA[Mf * 64 + k + K0];
    v2f b = *(const v2f*)&ldsW[((k + K0) >> 1) * 256 + nl * 2];
    c = __builtin_amdgcn_wmma_f32_16x16x4_f32(false, a, false, b,
                                              (short)0, c, false, false);
  }
  const int nOut  = nb0 + nl;
  const int mBase = m0 + ((lane >> 4) << 3);
  const float bias = bdt[nOut];
  #pragma unroll
  for (int r = 0; r < 8; ++r) {
    float v  = c[r] + bias;
    float sp = (v > 20.0f) ? v : log1pf(__expf(v));   // softplus
    dt[(size_t)(mBase + r) * D_INNER + nOut] = sp;
  }
}

// ---------------------------------------------------------------------------
// K3: selective scan. One lane per (d, n) pair: 65536 lanes = 2048 wave32s.
// Wave holds 2 channels x 16 states; y reduced over the 16 state lanes via
// shfl_xor. Recurrence fma is the only op on the sequential critical path;
// global_prefetch hides L2 latency of the t+4 broadcast loads.
// ---------------------------------------------------------------------------
__global__ __launch_bounds__(256)
void k_scan(const float* __restrict__ x, const float* __restrict__ xz,
            const float* __restrict__ dt, const float* __restrict__ A_log,
            const float* __restrict__ Dp, float* __restrict__ y) {
  const int tid  = threadIdx.x;
  const int wave = tid >> 5;
  const int lane = tid & 31;
  const int half = lane >> 4;
  const int n    = lane & 15;
  const int b    = blockIdx.x >> 7;                             // 0..1
  const int d    = ((blockIdx.x & 127) << 4) + wave * 2 + half; // 0..2047

  const float negA  = __expf(A_log[d * D_STATE + n]);  // neg_A = exp(A_log)
  const float dcoef = Dp[d];
  const size_t rowBase = (size_t)b * SEQ;

  float h = 0.0f;
  for (int t = 0; t < SEQ; ++t) {
    const size_t r = rowBase + t;
    if (n == 0 && t + 4 < SEQ) {                  // global_prefetch_b8
      __builtin_prefetch(&x [(r + 4) * D_INNER + d], 0, 0);
      __builtin_prefetch(&dt[(r + 4) * D_INNER + d], 0, 0);
    }
    const float u   = x [r * D_INNER + d];        // broadcast across 16 lanes
    const float dtv = dt[r * D_INNER + d];        // broadcast across 16 lanes
    const float Bn  = xz[r * NXZ + DT_RANK + n];
    const float Cn  = xz[r * NXZ + DT_RANK + D_STATE + n];

    const float abar = __expf(-dtv * negA);       // off critical path (pipelined)
    h = fmaf(abar, h, dtv * u * Bn);              // critical-path fma

    float p = h * Cn;                             // reduce over 16 state lanes
    p += __shfl_xor(p, 1);
    p += __shfl_xor(p, 2);
    p += __shfl_xor(p, 4);
    p += __shfl_xor(p, 8);
    if (n == 0) y[r * D_INNER + d] = fmaf(dcoef, u, p);
  }
}

// ---------------------------------------------------------------------------
extern "C" void kernel_launch(void* const* d_in, const int* in_sizes, int n_in,
                              void* d_out, int out_size, void* d_ws, size_t ws_size,
                              hipStream_t stream) {
  const float* x     = (const float*)d_in[0];
  const float* Wx    = (const float*)d_in[1];
  const float* Wdt   = (const float*)d_in[2];
  const float* bdt   = (const float*)d_in[3];
  const float* A_log = (const float*)d_in[4];
  const float* Dp    = (const float*)d_in[5];
  float* y = (float*)d_out;

  // workspace layout: xz [4096*96] floats, then dt [4096*2048] floats (~33.5 MB)
  float* xz = (float*)d_ws;
  float* dt = xz + (size_t)NROWS * NXZ;

  k_xz_gemm<<<dim3(NROWS / 16), dim3(192), 0, stream>>>(x, Wx, xz);
  k_dt_gemm<<<dim3(NROWS / 16, 16), dim3(256), 0, stream>>>(xz, Wdt, bdt, dt);
  k_scan  <<<dim3(BATCH * (D_INNER / 16)), dim3(256), 0, stream>>>(x, xz, dt, A_log, Dp, y);
}